// Fp8Unpadding_11948599018074
// MI455X (gfx1250) — compile-verified
//
#include <hip/hip_runtime.h>
#include <hip/hip_bf16.h>
#include <stdint.h>

// Fp8Unpadding: gather unpadded rows out of 256-row-padded groups.
//   out[r, :] = inp[src_row(r), :],  row width = 4096 f32 = 16 KB.
// Pure streaming copy -> bound by HBM (23.3 TB/s). ~400 MB traffic ~ 17 us.
//
// CDNA5 path: async global->LDS / LDS->global b128 DMA (ASYNCcnt), wave32,
// non-temporal cache hints (read-once / write-once streams).
// Block = 256 threads = 8 waves; each lane stores exactly the bytes it
// loaded, so a per-wave s_wait_asynccnt 0 is the only synchronization needed.

#define HIDDEN 4096
#define NGROUPS 8

__global__ void __launch_bounds__(256)
unpad_rows_async(const float* __restrict__ inp,
                 const int* __restrict__ m_splits,
                 float* __restrict__ out)
{
    __shared__ float buf[HIDDEN];   // 16 KB row staging buffer

    // --- uniform: load all splits up front (clauses into wide s_loads) -----
    int ms[NGROUPS];
    #pragma unroll
    for (int g = 0; g < NGROUPS; ++g) ms[g] = m_splits[g];

    // --- uniform, branch-free: map output row -> input row -----------------
    long long row     = (long long)blockIdx.x;
    long long out_off = 0;   // prefix sum of m_i over groups before ours
    long long in_off  = 0;   // prefix sum of pad256(m_i) over groups before ours
    bool found = false;
    #pragma unroll
    for (int g = 0; g < NGROUPS; ++g) {
        long long mi  = (long long)ms[g];
        bool adv      = !found && (row >= out_off + mi);  // group g is before ours
        if (adv) {
            out_off += mi;
            in_off  += (mi + 255) & ~255ll;
        }
        found = found || !adv;
    }
    long long src_row = in_off + (row - out_off);

    unsigned long long src = (unsigned long long)(inp + src_row * (long long)HIDDEN);
    unsigned long long dst = (unsigned long long)(out + row     * (long long)HIDDEN);

    // --- per-lane addresses ------------------------------------------------
    // thread t handles bytes {t*16 + p*4096 | p=0..3} of the row (b128 each).
    unsigned t    = threadIdx.x;
    unsigned voff = t * 16u;                          // byte offset within row
    // low 32 bits of the generic pointer == addrspace(3) byte offset
    unsigned lds  = (unsigned)(size_t)(&buf[0]) + voff;

    // --- async DMA: global -> LDS (4 x b128 per lane = 16 KB per block) ----
    // The 24-bit inst offset is added to BOTH the LDS and global addresses,
    // so one (lds, voff) pair covers all four passes.
    asm volatile(
        "global_load_async_to_lds_b128 %0, %1, %2 offset:0 th:TH_LOAD_NT\n\t"
        "global_load_async_to_lds_b128 %0, %1, %2 offset:4096 th:TH_LOAD_NT\n\t"
        "global_load_async_to_lds_b128 %0, %1, %2 offset:8192 th:TH_LOAD_NT\n\t"
        "global_load_async_to_lds_b128 %0, %1, %2 offset:12288 th:TH_LOAD_NT\n\t"
        "s_wait_asynccnt 0x0"
        :
        : "v"(lds), "v"(voff), "s"(src)
        : "memory");

    // --- async DMA: LDS -> global ------------------------------------------
    asm volatile(
        "global_store_async_from_lds_b128 %1, %0, %2 offset:0 th:TH_STORE_NT\n\t"
        "global_store_async_from_lds_b128 %1, %0, %2 offset:4096 th:TH_STORE_NT\n\t"
        "global_store_async_from_lds_b128 %1, %0, %2 offset:8192 th:TH_STORE_NT\n\t"
        "global_store_async_from_lds_b128 %1, %0, %2 offset:12288 th:TH_STORE_NT\n\t"
        "s_wait_asynccnt 0x0"
        :
        : "v"(lds), "v"(voff), "s"(dst)
        : "memory");
}

extern "C" void kernel_launch(void* const* d_in, const int* in_sizes, int n_in,
                              void* d_out, int out_size, void* d_ws, size_t ws_size,
                              hipStream_t stream)
{
    const float* inp      = (const float*)d_in[0];
    const int*   m_splits = (const int*)d_in[1];
    float*       out      = (float*)d_out;

    int n_rows = out_size / HIDDEN;   // 12166 for the reference splits
    if (n_rows <= 0) return;

    unpad_rows_async<<<dim3((unsigned)n_rows), dim3(256), 0, stream>>>(inp, m_splits, out);
}